// WNetLoss_89996744720871
// MI455X (gfx1250) — compile-verified
//
#include <hip/hip_runtime.h>

typedef __attribute__((ext_vector_type(2))) float v2f;
typedef __attribute__((ext_vector_type(8))) float v8f;

constexpr int Bn = 32;
constexpr int Hn = 1024;
constexpr int Wn = 1024;
constexpr int ROWS = 4;                 // rows per workgroup

struct Ws {
    double       ce_sum;                // sum of picked log-probs
    unsigned int cnt_trans;             // total count of x1 > x0
    unsigned int cnt_gt;                // total count of gt == 1
    int t_rmin[Bn], t_rmax[Bn], t_cmin[Bn], t_cmax[Bn];   // pred-mask bbox
    int g_rmin[Bn], g_rmax[Bn], g_cmin[Bn], g_cmax[Bn];   // gt-mask bbox
};

// wave32 lane exchange: ds_swizzle group-of-32 XOR pattern, single instruction,
// no LDS traffic, no bounds-check VALU (unlike __shfl_xor's ds_bpermute path).
#define SWZ(v, m) __builtin_amdgcn_ds_swizzle((v), (0x1f | ((m) << 10)))

__device__ inline float red_add_f(float v) {
    v += __int_as_float(SWZ(__float_as_int(v), 16));
    v += __int_as_float(SWZ(__float_as_int(v), 8));
    v += __int_as_float(SWZ(__float_as_int(v), 4));
    v += __int_as_float(SWZ(__float_as_int(v), 2));
    v += __int_as_float(SWZ(__float_as_int(v), 1));
    return v;
}
__device__ inline int red_add_i(int v) {
    v += SWZ(v, 16); v += SWZ(v, 8); v += SWZ(v, 4); v += SWZ(v, 2); v += SWZ(v, 1);
    return v;
}
__device__ inline int red_or_i(int v) {
    v |= SWZ(v, 16); v |= SWZ(v, 8); v |= SWZ(v, 4); v |= SWZ(v, 2); v |= SWZ(v, 1);
    return v;
}
__device__ inline int red_min_i(int v) {
    v = min(v, SWZ(v, 16)); v = min(v, SWZ(v, 8)); v = min(v, SWZ(v, 4));
    v = min(v, SWZ(v, 2));  v = min(v, SWZ(v, 1));
    return v;
}
__device__ inline int red_max_i(int v) {
    v = max(v, SWZ(v, 16)); v = max(v, SWZ(v, 8)); v = max(v, SWZ(v, 4));
    v = max(v, SWZ(v, 2));  v = max(v, SWZ(v, 1));
    return v;
}
// second stage: 8 wave-partials live in lanes 0..7
__device__ inline float red8_add_f(float v) {
    v += __int_as_float(SWZ(__float_as_int(v), 4));
    v += __int_as_float(SWZ(__float_as_int(v), 2));
    v += __int_as_float(SWZ(__float_as_int(v), 1));
    return v;
}
__device__ inline int red8_add_i(int v) { v += SWZ(v,4); v += SWZ(v,2); v += SWZ(v,1); return v; }
__device__ inline int red8_or_i(int v)  { v |= SWZ(v,4); v |= SWZ(v,2); v |= SWZ(v,1); return v; }
__device__ inline int red8_min_i(int v) { v = min(v,SWZ(v,4)); v = min(v,SWZ(v,2)); v = min(v,SWZ(v,1)); return v; }
__device__ inline int red8_max_i(int v) { v = max(v,SWZ(v,4)); v = max(v,SWZ(v,2)); v = max(v,SWZ(v,1)); return v; }

__global__ void init_ws_kernel(Ws* __restrict__ ws) {
    const int t = threadIdx.x;
    if (t == 0) { ws->ce_sum = 0.0; ws->cnt_trans = 0u; ws->cnt_gt = 0u; }
    if (t < Bn) {
        ws->t_rmin[t] = Hn; ws->t_rmax[t] = -1; ws->t_cmin[t] = Wn; ws->t_cmax[t] = -1;
        ws->g_rmin[t] = Hn; ws->g_rmax[t] = -1; ws->g_cmin[t] = Wn; ws->g_cmax[t] = -1;
    }
}

// One workgroup per (sample, 4 rows). 256 threads; each thread streams 12
// independent b128 loads (x0/x1/gt for 4 rows), then one reduction tail.
__global__ __launch_bounds__(256) void row_reduce_kernel(const float* __restrict__ x,
                                                         const int*   __restrict__ gt,
                                                         Ws* __restrict__ ws) {
    const int blk   = blockIdx.x;              // 0 .. 8191
    const int b     = blk >> 8;                // 256 blocks per sample
    const int hbase = (blk & 255) << 2;        // first of 4 rows

    const float* x0base = x  + ((size_t)(2 * b)     * Hn + hbase) * Wn;
    const float* x1base = x  + ((size_t)(2 * b + 1) * Hn + hbase) * Wn;
    const int*   gbase  = gt + ((size_t)b           * Hn + hbase) * Wn;

    const int t = threadIdx.x;

    float4 a0[ROWS], a1[ROWS];
    int4   gv[ROWS];
#pragma unroll
    for (int r = 0; r < ROWS; ++r) {           // issue all 12 loads up front
        a0[r] = ((const float4*)(x0base + (size_t)r * Wn))[t];
        a1[r] = ((const float4*)(x1base + (size_t)r * Wn))[t];
        gv[r] = ((const int4*)  (gbase  + (size_t)r * Wn))[t];
    }

    float psum = 0.0f;
    int   cnt  = 0;                            // [15:0]=trans count, [31:16]=gt count
    int   any  = 0;                            // bits[3:0]=trans row-any, [7:4]=gt row-any
    int cminT = 1 << 30, cmaxT = -1, cminG = 1 << 30, cmaxG = -1;
    const int wbase = t << 2;

#pragma unroll
    for (int r = 0; r < ROWS; ++r) {
#pragma unroll
        for (int j = 0; j < 4; ++j) {
            const float e0 = (&a0[r].x)[j];
            const float e1 = (&a1[r].x)[j];
            const int   gg = (&gv[r].x)[j];
            const float d  = e1 - e0;
            // picked = -softplus(s), s = +d (gt=0) or -d (gt=1); stable form
            const float s  = gg ? -d : d;
            psum -= fmaxf(s, 0.0f) + __logf(1.0f + __expf(-fabsf(s)));
            const int w = wbase + j;
            if (d > 0.0f) { cnt += 1;       any |= 1 << r;       cminT = min(cminT, w); cmaxT = max(cmaxT, w); }
            if (gg)       { cnt += 1 << 16; any |= 1 << (4 + r); cminG = min(cminG, w); cmaxG = max(cmaxG, w); }
        }
    }

    // intra-wave reductions (ds_swizzle, 1 instr/step)
    psum  = red_add_f(psum);
    cnt   = red_add_i(cnt);
    any   = red_or_i(any);
    cminT = red_min_i(cminT); cmaxT = red_max_i(cmaxT);
    cminG = red_min_i(cminG); cmaxG = red_max_i(cmaxG);

    __shared__ float sps[8];
    __shared__ int   scnt[8], sany[8], sminT[8], smaxT[8], sminG[8], smaxG[8];
    const int wid = t >> 5;
    if ((t & 31) == 0) {
        sps[wid] = psum; scnt[wid] = cnt; sany[wid] = any;
        sminT[wid] = cminT; smaxT[wid] = cmaxT; sminG[wid] = cminG; smaxG[wid] = cmaxG;
    }
    __syncthreads();

    if (t < 32) {
        const bool v = (t < 8);
        psum  = v ? sps[t]  : 0.0f;
        cnt   = v ? scnt[t] : 0;
        any   = v ? sany[t] : 0;
        cminT = v ? sminT[t] : (1 << 30);
        cmaxT = v ? smaxT[t] : -1;
        cminG = v ? sminG[t] : (1 << 30);
        cmaxG = v ? smaxG[t] : -1;

        psum  = red8_add_f(psum);
        cnt   = red8_add_i(cnt);
        any   = red8_or_i(any);
        cminT = red8_min_i(cminT); cmaxT = red8_max_i(cmaxT);
        cminG = red8_min_i(cminG); cmaxG = red8_max_i(cmaxG);

        if (t == 0) {
            atomicAdd(&ws->ce_sum, (double)psum);
            const int ct = cnt & 0xFFFF;
            const int cg = (cnt >> 16) & 0xFFFF;
            if (ct) atomicAdd(&ws->cnt_trans, (unsigned)ct);
            if (cg) atomicAdd(&ws->cnt_gt,    (unsigned)cg);
            const int bitsT = any & 0xF;
            const int bitsG = (any >> 4) & 0xF;
            if (bitsT) {
                atomicMin(&ws->t_rmin[b], hbase + (__ffs(bitsT) - 1));
                atomicMax(&ws->t_rmax[b], hbase + (31 - __clz(bitsT)));
                atomicMin(&ws->t_cmin[b], cminT);
                atomicMax(&ws->t_cmax[b], cmaxT);
            }
            if (bitsG) {
                atomicMin(&ws->g_rmin[b], hbase + (__ffs(bitsG) - 1));
                atomicMax(&ws->g_rmax[b], hbase + (31 - __clz(bitsG)));
                atomicMin(&ws->g_cmin[b], cminG);
                atomicMax(&ws->g_cmax[b], cmaxG);
            }
        }
    }
}

__device__ inline float sqf(float v) { return v * v; }

// One wave; lane b handles sample b. Final B=32 reduction on the matrix pipe:
// D = A(16x4) x ones(4x16), A[m,k] = {tc, tl} per lane -> exact f32 row sums;
// per-lane sum of D[0..7] + cross-half swizzle = sum over 32 lanes of (tc+tl).
__global__ void finalize_kernel(const Ws* __restrict__ ws, float* __restrict__ out) {
    const int b = threadIdx.x;   // 0..31, EXEC all ones (WMMA requirement)

    const float trmin = (float)ws->t_rmin[b], trmax = (float)ws->t_rmax[b];
    const float tcmin = (float)ws->t_cmin[b], tcmax = (float)ws->t_cmax[b];
    const float grmin = (float)ws->g_rmin[b], grmax = (float)ws->g_rmax[b];
    const float gcmin = (float)ws->g_cmin[b], gcmax = (float)ws->g_cmax[b];

    const float invW = 1.0f / (float)Wn, invH = 1.0f / (float)Hn;
    // (faithful to reference: row coords / W, col coords / H)
    const float xcx = (trmin + trmax) * 0.5f * invW;
    const float xcy = (tcmin + tcmax) * 0.5f * invH;
    const float xln = sqf((trmax - trmin) * invW) + sqf((tcmax - tcmin) * invH);
    const float gcx = (grmin + grmax) * 0.5f * invW;
    const float gcy = (gcmin + gcmax) * 0.5f * invH;
    const float gln = sqf((grmax - grmin) * invW) + sqf((gcmax - gcmin) * invH);

    const float tc = sqf(xcx - gcx) + sqf(xcy - gcy);   // per-sample center dist
    const float tl = sqf(xln - gln);                    // per-sample length dist

    int ok = (ws->t_rmax[b] >= 0) && (ws->g_rmax[b] >= 0);
    ok = (red_or_i(ok ^ 1) == 0) ? 1 : 0;               // all 32 samples nonempty

    v2f A;  A[0] = tc;    A[1] = tl;     // A 16x4 f32: lane m -> K0/K1, lane m+16 -> K2/K3
    v2f Bm; Bm[0] = 1.0f; Bm[1] = 1.0f;  // B 4x16 all-ones
    v8f C = {};
    v8f D = __builtin_amdgcn_wmma_f32_16x16x4_f32(
        /*neg_a=*/false, A, /*neg_b=*/false, Bm,
        /*c_mod=*/(short)0, C, /*reuse_a=*/false, /*reuse_b=*/false);

    float s = D[0] + D[1] + D[2] + D[3] + D[4] + D[5] + D[6] + D[7];
    s += __int_as_float(SWZ(__float_as_int(s), 16));    // M=0..7 half + M=8..15 half

    if (b == 0) {
        const float area = (float)((size_t)Bn * Hn * Wn);
        const float cel  = -(float)(ws->ce_sum / (double)((size_t)Bn * Hn * Wn));
        const float diff = ((float)ws->cnt_trans - (float)ws->cnt_gt) / area;
        const float area_loss = diff * diff;
        const float dist = s / (float)Bn;               // dist_c + dist_length
        out[0] = cel + area_loss + 0.5f * (ok ? 1.0f : 0.0f) * dist;
    }
}

extern "C" void kernel_launch(void* const* d_in, const int* in_sizes, int n_in,
                              void* d_out, int out_size, void* d_ws, size_t ws_size,
                              hipStream_t stream) {
    (void)in_sizes; (void)n_in; (void)out_size; (void)ws_size;
    const float* x  = (const float*)d_in[0];
    const int*   gt = (const int*)d_in[1];
    Ws*    ws  = (Ws*)d_ws;
    float* out = (float*)d_out;

    init_ws_kernel<<<1, 32, 0, stream>>>(ws);
    row_reduce_kernel<<<(Bn * Hn) / ROWS, 256, 0, stream>>>(x, gt, ws);
    finalize_kernel<<<1, 32, 0, stream>>>(ws, out);
}